// GPRGNN_24481313587859
// MI455X (gfx1250) — compile-verified
//
#include <hip/hip_runtime.h>
#include <stdint.h>

typedef float v2f __attribute__((ext_vector_type(2)));
typedef float v8f __attribute__((ext_vector_type(8)));

#define IN_DIM  512
#define HIDDEN  512
#define OUT_DIM 128

// ---------------------------------------------------------------------------
// GEMM1: h[n,512] = relu(x[n,512] @ Win^T + b), Win is [512,512] row-major.
// One wave computes a 16(M) x 64(N) tile with V_WMMA_F32_16X16X4_F32.
// Block = 128 threads (4 waves), grid = (n/16, 2) -> 8 N-groups of 64.
// ---------------------------------------------------------------------------
__global__ __launch_bounds__(128) void gemm_in_relu(
    const float* __restrict__ x, const float* __restrict__ W,
    const float* __restrict__ bias, float* __restrict__ h)
{
    const int wave = threadIdx.x >> 5;
    const int lane = threadIdx.x & 31;
    const int lo = lane & 15;
    const int hi = lane >> 4;
    const int m0 = blockIdx.x << 4;
    const int n0 = ((blockIdx.y << 2) + wave) << 6;   // 64 cols per wave

    const float* ap  = x + (size_t)(m0 + lo) * IN_DIM + 2 * hi;
    const float* bp0 = W + (size_t)(n0 +  0 + lo) * IN_DIM + 2 * hi;
    const float* bp1 = W + (size_t)(n0 + 16 + lo) * IN_DIM + 2 * hi;
    const float* bp2 = W + (size_t)(n0 + 32 + lo) * IN_DIM + 2 * hi;
    const float* bp3 = W + (size_t)(n0 + 48 + lo) * IN_DIM + 2 * hi;

    v8f c0 = {}, c1 = {}, c2 = {}, c3 = {};
    for (int k0 = 0; k0 < IN_DIM; k0 += 4) {
        v2f a  = *(const v2f*)(ap  + k0);
        v2f b0 = *(const v2f*)(bp0 + k0);
        v2f b1 = *(const v2f*)(bp1 + k0);
        v2f b2 = *(const v2f*)(bp2 + k0);
        v2f b3 = *(const v2f*)(bp3 + k0);
        c0 = __builtin_amdgcn_wmma_f32_16x16x4_f32(false, a, false, b0, (short)0, c0, false, false);
        c1 = __builtin_amdgcn_wmma_f32_16x16x4_f32(false, a, false, b1, (short)0, c1, false, false);
        c2 = __builtin_amdgcn_wmma_f32_16x16x4_f32(false, a, false, b2, (short)0, c2, false, false);
        c3 = __builtin_amdgcn_wmma_f32_16x16x4_f32(false, a, false, b3, (short)0, c3, false, false);
    }

    const float bb0 = bias[n0 +  0 + lo];
    const float bb1 = bias[n0 + 16 + lo];
    const float bb2 = bias[n0 + 32 + lo];
    const float bb3 = bias[n0 + 48 + lo];
#pragma unroll
    for (int v = 0; v < 8; ++v) {
        const int row = m0 + v + (hi << 3);
        float* hr = h + (size_t)row * HIDDEN + n0 + lo;
        float t;
        t = c0[v] + bb0; hr[ 0] = t > 0.f ? t : 0.f;
        t = c1[v] + bb1; hr[16] = t > 0.f ? t : 0.f;
        t = c2[v] + bb2; hr[32] = t > 0.f ? t : 0.f;
        t = c3[v] + bb3; hr[48] = t > 0.f ? t : 0.f;
    }
}

// ---------------------------------------------------------------------------
// GEMM2: out[n,128] = acc[n,512] @ Wout^T + b, Wout is [128,512] row-major.
// Block = 64 threads (2 waves), each wave a 16x64 tile; grid = (n/16).
// ---------------------------------------------------------------------------
__global__ __launch_bounds__(64) void gemm_out_bias(
    const float* __restrict__ acc, const float* __restrict__ W,
    const float* __restrict__ bias, float* __restrict__ out)
{
    const int wave = threadIdx.x >> 5;
    const int lane = threadIdx.x & 31;
    const int lo = lane & 15;
    const int hi = lane >> 4;
    const int m0 = blockIdx.x << 4;
    const int n0 = wave << 6;

    const float* ap  = acc + (size_t)(m0 + lo) * HIDDEN + 2 * hi;
    const float* bp0 = W + (size_t)(n0 +  0 + lo) * HIDDEN + 2 * hi;
    const float* bp1 = W + (size_t)(n0 + 16 + lo) * HIDDEN + 2 * hi;
    const float* bp2 = W + (size_t)(n0 + 32 + lo) * HIDDEN + 2 * hi;
    const float* bp3 = W + (size_t)(n0 + 48 + lo) * HIDDEN + 2 * hi;

    v8f c0 = {}, c1 = {}, c2 = {}, c3 = {};
    for (int k0 = 0; k0 < HIDDEN; k0 += 4) {
        v2f a  = *(const v2f*)(ap  + k0);
        v2f b0 = *(const v2f*)(bp0 + k0);
        v2f b1 = *(const v2f*)(bp1 + k0);
        v2f b2 = *(const v2f*)(bp2 + k0);
        v2f b3 = *(const v2f*)(bp3 + k0);
        c0 = __builtin_amdgcn_wmma_f32_16x16x4_f32(false, a, false, b0, (short)0, c0, false, false);
        c1 = __builtin_amdgcn_wmma_f32_16x16x4_f32(false, a, false, b1, (short)0, c1, false, false);
        c2 = __builtin_amdgcn_wmma_f32_16x16x4_f32(false, a, false, b2, (short)0, c2, false, false);
        c3 = __builtin_amdgcn_wmma_f32_16x16x4_f32(false, a, false, b3, (short)0, c3, false, false);
    }

    const float bb0 = bias[n0 +  0 + lo];
    const float bb1 = bias[n0 + 16 + lo];
    const float bb2 = bias[n0 + 32 + lo];
    const float bb3 = bias[n0 + 48 + lo];
#pragma unroll
    for (int v = 0; v < 8; ++v) {
        const int row = m0 + v + (hi << 3);
        float* orow = out + (size_t)row * OUT_DIM + n0 + lo;
        orow[ 0] = c0[v] + bb0;
        orow[16] = c1[v] + bb1;
        orow[32] = c2[v] + bb2;
        orow[48] = c3[v] + bb3;
    }
}

// ---------------------------------------------------------------------------
// Small helpers
// ---------------------------------------------------------------------------
__global__ void softmax_small(const float* __restrict__ g, float* __restrict__ w, int n)
{
    if (blockIdx.x == 0 && threadIdx.x == 0) {
        float m = g[0];
        for (int i = 1; i < n; ++i) m = fmaxf(m, g[i]);
        float s = 0.f;
        for (int i = 0; i < n; ++i) { float e = expf(g[i] - m); w[i] = e; s += e; }
        float inv = 1.f / s;
        for (int i = 0; i < n; ++i) w[i] *= inv;
    }
}

__global__ void fill_ones(float* __restrict__ p, int n)
{
    int i = blockIdx.x * blockDim.x + threadIdx.x;
    if (i < n) p[i] = 1.0f;
}

__global__ void deg_edges(const int* __restrict__ rows, float* __restrict__ deg, int E)
{
    int i = blockIdx.x * blockDim.x + threadIdx.x;
    if (i < E)
        __hip_atomic_fetch_add(&deg[rows[i]], 1.0f, __ATOMIC_RELAXED, __HIP_MEMORY_SCOPE_AGENT);
}

__global__ void make_dis(const float* __restrict__ deg, float* __restrict__ dis,
                         float* __restrict__ dis2, int n)
{
    int i = blockIdx.x * blockDim.x + threadIdx.x;
    if (i < n) {
        float d = fmaxf(deg[i], 1.0f);
        float r = 1.0f / sqrtf(d);
        dis[i]  = r;
        dis2[i] = r * r;
    }
}

// acc = w[0]*h ; nxt = dis2*h  (self-loop init for step 1).  q = n*512/4.
__global__ void init_acc_self(const float4* __restrict__ h, const float* __restrict__ w,
                              const float* __restrict__ dis2, float4* __restrict__ acc,
                              float4* __restrict__ nxt, int q)
{
    int i = blockIdx.x * blockDim.x + threadIdx.x;
    if (i >= q) return;
    const float w0 = w[0];
    const float d2 = dis2[i >> 7];          // 128 float4 per node row
    float4 hv = h[i];
    acc[i] = make_float4(w0 * hv.x, w0 * hv.y, w0 * hv.z, w0 * hv.w);
    nxt[i] = make_float4(d2 * hv.x, d2 * hv.y, d2 * hv.z, d2 * hv.w);
}

// dst[r] += dis[r]*dis[c]*src[c]  — 128 threads/edge, float4 per thread,
// coalesced 2KB gather of the source row, 4x f32 atomic scatter-add.
__global__ __launch_bounds__(256) void prop_edges(
    const int* __restrict__ rows, const int* __restrict__ cols,
    const float* __restrict__ dis, const float* __restrict__ src,
    float* __restrict__ dst, int E)
{
    int tid = blockIdx.x * 256 + threadIdx.x;
    int e = tid >> 7;
    if (e >= E) return;
    int f = (tid & 127) << 2;
    int r = rows[e];
    int c = cols[e];
    float coef = dis[r] * dis[c];
    float4 s = *(const float4*)(src + (size_t)c * HIDDEN + f);
    float* d = dst + (size_t)r * HIDDEN + f;
    __hip_atomic_fetch_add(d + 0, coef * s.x, __ATOMIC_RELAXED, __HIP_MEMORY_SCOPE_AGENT);
    __hip_atomic_fetch_add(d + 1, coef * s.y, __ATOMIC_RELAXED, __HIP_MEMORY_SCOPE_AGENT);
    __hip_atomic_fetch_add(d + 2, coef * s.z, __ATOMIC_RELAXED, __HIP_MEMORY_SCOPE_AGENT);
    __hip_atomic_fetch_add(d + 3, coef * s.w, __ATOMIC_RELAXED, __HIP_MEMORY_SCOPE_AGENT);
}

// acc += w[k]*cur ; if(doInit) nxt = dis2*cur (self-loop init for next step).
__global__ void finish_step(const float4* __restrict__ cur, float4* __restrict__ acc,
                            float4* __restrict__ nxt, const float* __restrict__ dis2,
                            const float* __restrict__ w, int k, int doInit, int q)
{
    int i = blockIdx.x * blockDim.x + threadIdx.x;
    if (i >= q) return;
    const float wk = w[k];
    float4 c = cur[i];
    float4 a = acc[i];
    a.x += wk * c.x; a.y += wk * c.y; a.z += wk * c.z; a.w += wk * c.w;
    acc[i] = a;
    if (doInit) {
        const float d2 = dis2[i >> 7];
        nxt[i] = make_float4(d2 * c.x, d2 * c.y, d2 * c.z, d2 * c.w);
    }
}

// ---------------------------------------------------------------------------
extern "C" void kernel_launch(void* const* d_in, const int* in_sizes, int n_in,
                              void* d_out, int out_size, void* d_ws, size_t ws_size,
                              hipStream_t stream)
{
    const float* x    = (const float*)d_in[0];
    const int*   ei   = (const int*)  d_in[1];   // int32 (JAX x64 disabled)
    const float* Win  = (const float*)d_in[2];
    const float* bin  = (const float*)d_in[3];
    const float* Wout = (const float*)d_in[4];
    const float* bout = (const float*)d_in[5];
    const float* gpr  = (const float*)d_in[6];

    const int NN = in_sizes[0] / IN_DIM;         // 50000
    const int E  = in_sizes[1] / 2;              // 1,600,000
    const int K1 = in_sizes[6];                  // 11
    const int* rows = ei;
    const int* cols = ei + E;

    // workspace carve-out (256B aligned)
    uintptr_t p = ((uintptr_t)d_ws + 255) & ~(uintptr_t)255;
    auto bump = [&](size_t bytes) {
        void* r = (void*)p;
        p = (p + bytes + 255) & ~(uintptr_t)255;
        return r;
    };
    float* w    = (float*)bump(64);
    float* deg  = (float*)bump((size_t)NN * 4);
    float* dis  = (float*)bump((size_t)NN * 4);
    float* dis2 = (float*)bump((size_t)NN * 4);
    float* bufA = (float*)bump((size_t)NN * HIDDEN * 4);
    float* bufB = (float*)bump((size_t)NN * HIDDEN * 4);
    float* acc  = (float*)bump((size_t)NN * HIDDEN * 4);
    (void)ws_size; (void)n_in; (void)out_size;

    const int q = NN * HIDDEN / 4;               // float4 count

    softmax_small<<<1, 32, 0, stream>>>(gpr, w, K1);
    fill_ones<<<(NN + 255) / 256, 256, 0, stream>>>(deg, NN);
    deg_edges<<<(E + 255) / 256, 256, 0, stream>>>(rows, deg, E);
    make_dis<<<(NN + 255) / 256, 256, 0, stream>>>(deg, dis, dis2, NN);

    dim3 g1(NN / 16, 2);
    gemm_in_relu<<<g1, 128, 0, stream>>>(x, Win, bin, bufA);

    init_acc_self<<<(q + 255) / 256, 256, 0, stream>>>(
        (const float4*)bufA, w, dis2, (float4*)acc, (float4*)bufB, q);

    float* src = bufA;   // h_prop of current step
    float* dst = bufB;   // pre-initialized with self-loop term
    const unsigned eblocks = (unsigned)(((size_t)E * 128 + 255) / 256);
    for (int k = 1; k < K1; ++k) {
        prop_edges<<<eblocks, 256, 0, stream>>>(rows, cols, dis, src, dst, E);
        finish_step<<<(q + 255) / 256, 256, 0, stream>>>(
            (const float4*)dst, (float4*)acc, (float4*)src, dis2, w, k,
            (k < K1 - 1) ? 1 : 0, q);
        float* t = src; src = dst; dst = t;
    }

    gemm_out_bias<<<NN / 16, 64, 0, stream>>>(acc, Wout, bout, (float*)d_out);
}